// WN_7121055776850
// MI455X (gfx1250) — compile-verified
//
#include <hip/hip_runtime.h>
#include <hip/hip_bf16.h>

// ---------------------------------------------------------------------------
// Problem constants (from the reference):
//   x : [B=8, C=128, T=32768] f32      W : [L=8, 128, 128, KS=2] f32
//   b : [L=8, 128] f32                 out: [8, 128, 32768] f32
// Only kernel tap 0 contributes, so the whole module is ONE affine map:
//   out = (W7*...*W0) @ x + c,  c = b7 + W7*(b6 + W6*(...))
// Roofline: 256 MB of traffic / 23.3 TB/s ~= 11 us; 8.6 GFLOP of f32 GEMM on
// v_wmma_f32_16x16x4_f32 lands in the same ballpark -> near-balanced, full f32.
// ---------------------------------------------------------------------------

#define NLAY 8
#define CCH  128
#define TT   32768
#define NB   8
#define KS   2

#define PAD  144   // LDS row stride (dwords): 144 % 64 = 16, 2 rows -> +32 banks

typedef __attribute__((ext_vector_type(2))) float v2f;
typedef __attribute__((ext_vector_type(4))) float v4f;
typedef __attribute__((ext_vector_type(8))) float v8f;

// async-copy builtin wants int4 pointers (per hipcc diagnostic)
typedef int v4i_vs __attribute__((vector_size(4 * sizeof(int))));
typedef __attribute__((address_space(1))) v4i_vs* g_v4i_ptr;
typedef __attribute__((address_space(3))) v4i_vs* l_v4i_ptr;

#if defined(__has_builtin)
#if __has_builtin(__builtin_amdgcn_global_load_async_to_lds_b128)
#define HAVE_ASYNC_LDS 1
#endif
#endif

// ---------------------------------------------------------------------------
// Kernel 1: collapse the 8-layer chain into Meff[128x128] + ceff[128] (d_ws).
// Single workgroup, 8 wave32s, WMMA f32 for the chain products. LDS holds the
// running product Ms (padded); the new product lives in accumulator registers
// between the read and write phases, so one buffer suffices.
// ---------------------------------------------------------------------------
__global__ __launch_bounds__(256) void wn_collapse(
    const float* __restrict__ W,   // [L,128,128,2]
    const float* __restrict__ bv,  // [L,128]
    float* __restrict__ Meff,      // [128*128]
    float* __restrict__ ceff)      // [128]
{
    extern __shared__ float sm[];         // 128*PAD + 128 floats
    float* Ms = sm;
    float* cs = sm + CCH * PAD;

    const int tid  = threadIdx.x;
    const int wave = tid >> 5;
    const int lane = tid & 31;
    const int hi   = lane >> 4;           // 0: lanes 0-15, 1: lanes 16-31
    const int l15  = lane & 15;
    const int arow = wave * 16 + l15;

    // ---- init Ms = W0[:, :, 0], cs = b0 ----
    for (int j = 0; j < 64; ++j) {
        int idx = tid + j * 256;          // 16384 entries
        int r = idx >> 7, cidx = idx & 127;
        Ms[r * PAD + cidx] = W[((size_t)r * CCH + cidx) * KS];
    }
    if (tid < CCH) cs[tid] = bv[tid];
    __syncthreads();

    for (int l = 1; l < NLAY; ++l) {
        // A fragments for W_l rows [16*wave, 16*wave+16): i-stride is KS=2
        const float* Wl = W + (((size_t)l * CCH + arow) * CCH) * KS;
        v2f afrag[32];
        #pragma unroll
        for (int k = 0; k < 32; ++k) {
            int i0 = 4 * k + 2 * hi;
            afrag[k].x = Wl[(size_t)i0 * KS];
            afrag[k].y = Wl[(size_t)(i0 + 1) * KS];
        }

        // Mnew = W_l @ Ms : 8 column tiles of 16, K = 128 (32 wmma steps each)
        v8f accs[8];
        #pragma unroll 1
        for (int n = 0; n < 8; ++n) {
            v8f acc = {};
            const float* bs = Ms + n * 16 + l15;
            #pragma unroll
            for (int k = 0; k < 32; ++k) {
                v2f bfrag;
                bfrag.x = bs[(size_t)(4 * k + 2 * hi) * PAD];
                bfrag.y = bs[(size_t)(4 * k + 1 + 2 * hi) * PAD];
                acc = __builtin_amdgcn_wmma_f32_16x16x4_f32(
                        false, afrag[k], false, bfrag, (short)0, acc,
                        false, false);
            }
            accs[n] = acc;
        }

        // bias chain: cnew = W_l @ cs + b_l  (cheap, 128 lanes of VALU)
        float cacc = 0.0f;
        if (tid < CCH) {
            const float* Wr = W + (((size_t)l * CCH + tid) * CCH) * KS;
            for (int i = 0; i < CCH; ++i)
                cacc = fmaf(Wr[(size_t)i * KS], cs[i], cacc);
            cacc += bv[(size_t)l * CCH + tid];
        }
        __syncthreads();                  // everyone done reading Ms/cs

        #pragma unroll
        for (int n = 0; n < 8; ++n)
            #pragma unroll
            for (int v = 0; v < 8; ++v) {
                int r = wave * 16 + v + 8 * hi;
                Ms[r * PAD + n * 16 + l15] = accs[n][v];
            }
        if (tid < CCH) cs[tid] = cacc;
        __syncthreads();
    }

    // ---- export (strip padding) ----
    for (int j = 0; j < 64; ++j) {
        int idx = tid + j * 256;
        int r = idx >> 7, cidx = idx & 127;
        Meff[idx] = Ms[r * PAD + cidx];
    }
    if (tid < CCH) ceff[tid] = cs[tid];
}

// ---------------------------------------------------------------------------
// Kernel 2: out = Meff @ x + ceff over 262144 columns.
// Block = 256 threads (8 wave32s); wave w owns output rows [16w, 16w+16).
// A-fragments live in 64 VGPRs, loaded ONCE per block. The 128x128 x-tile is
// staged into padded LDS with GLOBAL_LOAD_ASYNC_TO_LDS_B128 (ASYNCcnt path,
// overlapped with the A-fragment loads) when the builtin exists; otherwise a
// b128 load/ds_store fallback. Then 8 tiles x 32 v_wmma_f32_16x16x4_f32.
// ---------------------------------------------------------------------------
__global__ __launch_bounds__(256) void wn_gemm(
    const float* __restrict__ x,
    const float* __restrict__ Meff,
    const float* __restrict__ ceff,
    float* __restrict__ out)
{
    extern __shared__ float smem[];       // 128 * PAD floats
    const int tid  = threadIdx.x;
    const int wave = tid >> 5;
    const int lane = tid & 31;
    const int hi   = lane >> 4;
    const int l15  = lane & 15;

    // column block: 128 columns; T = 32768 is divisible by 128, so each
    // block stays inside one batch image.
    const long nc0 = (long)blockIdx.x * 128;
    const int  b   = (int)(nc0 >> 15);    // / T
    const int  t0  = (int)(nc0 & (TT - 1));
    const float* xb = x   + (size_t)b * CCH * TT + t0;
    float*       ob = out + (size_t)b * CCH * TT + t0;

    // ---- kick off x-tile staging FIRST so the DMA overlaps A loads ----
#if defined(HAVE_ASYNC_LDS)
    #pragma unroll
    for (int j = 0; j < 16; ++j) {
        int f  = tid + j * 256;           // 4096 float4's
        int r  = f >> 5;                  // 32 float4 per row
        int c4 = f & 31;
        __builtin_amdgcn_global_load_async_to_lds_b128(
            (g_v4i_ptr)(xb + (size_t)r * TT + c4 * 4),
            (l_v4i_ptr)(smem + r * PAD + c4 * 4),
            0, 0);
    }
#endif

    // ---- persistent A fragments: Meff rows [16*wave, +16), all K ----
    const int arow = wave * 16 + l15;
    const float* Mr = Meff + (size_t)arow * CCH;
    v2f afrag[32];
    #pragma unroll
    for (int k = 0; k < 32; ++k)
        afrag[k] = *(const v2f*)(Mr + 4 * k + 2 * hi);   // 8B aligned

    float bias[8];
    #pragma unroll
    for (int v = 0; v < 8; ++v)
        bias[v] = ceff[wave * 16 + v + 8 * hi];

#if defined(HAVE_ASYNC_LDS)
#if __has_builtin(__builtin_amdgcn_s_wait_asynccnt)
    __builtin_amdgcn_s_wait_asynccnt(0);
#else
    asm volatile("s_wait_asynccnt 0" ::: "memory");
#endif
#else
    // fallback: VGPR round-trip staging, b128 coalesced
    #pragma unroll
    for (int j = 0; j < 16; ++j) {
        int f   = tid + j * 256;
        int r   = f >> 5;
        int c4  = f & 31;
        v4f val = *(const v4f*)(xb + (size_t)r * TT + c4 * 4);
        *(v4f*)(smem + r * PAD + c4 * 4) = val;
    }
#endif
    __syncthreads();

    // ---- compute: 8 column tiles of 16, K = 128 ----
    #pragma unroll 1
    for (int n = 0; n < 8; ++n) {
        v8f acc;
        #pragma unroll
        for (int v = 0; v < 8; ++v) acc[v] = bias[v];

        const float* bs = smem + n * 16 + l15;
        #pragma unroll
        for (int k = 0; k < 32; ++k) {
            v2f bfrag;
            bfrag.x = bs[(size_t)(4 * k + 2 * hi) * PAD];
            bfrag.y = bs[(size_t)(4 * k + 1 + 2 * hi) * PAD];
            acc = __builtin_amdgcn_wmma_f32_16x16x4_f32(
                    false, afrag[k], false, bfrag, (short)0, acc,
                    false, false);
        }

        #pragma unroll
        for (int v = 0; v < 8; ++v) {
            int r = wave * 16 + v + 8 * hi;
            ob[(size_t)r * TT + n * 16 + l15] = acc[v];
        }
    }
}

// ---------------------------------------------------------------------------
extern "C" void kernel_launch(void* const* d_in, const int* in_sizes, int n_in,
                              void* d_out, int out_size, void* d_ws, size_t ws_size,
                              hipStream_t stream) {
    const float* x  = (const float*)d_in[0];   // [8,128,32768]
    const float* W  = (const float*)d_in[1];   // [8,128,128,2]
    const float* bv = (const float*)d_in[2];   // [8,128]
    float* out = (float*)d_out;

    float* Meff = (float*)d_ws;                // 16384 floats
    float* ceff = Meff + CCH * CCH;            // 128 floats

    const size_t lds1 = (size_t)(CCH * PAD + CCH) * sizeof(float); // 74240 B
    const size_t lds2 = (size_t)(CCH * PAD) * sizeof(float);       // 73728 B

    wn_collapse<<<1, 256, lds1, stream>>>(W, bv, Meff, ceff);

    const int ncols_blocks = (NB * TT) / 128;  // 2048
    wn_gemm<<<ncols_blocks, 256, lds2, stream>>>(x, Meff, ceff, out);
}